// HebbianChimera_12996571038028
// MI455X (gfx1250) — compile-verified
//
#include <hip/hip_runtime.h>
#include <math.h>

// ---------------------------------------------------------------------------
// HebbianChimera reservoir on MI455X (gfx1250).
// Single persistent workgroup (16 wave32 = 512 threads) on one WGP:
//   h [256x128] f32 (128 KB) + W [128x128] f32 (64 KB) + staging live in the
//   320 KB LDS for all 2000 sequential steps. Both GEMMs (h@W and h'^T h)
//   use V_WMMA_F32_16X16X4_F32 so precision matches the f32 reference.
//   Per-step x[:,t,:] staging (256 rows x 8B, 16KB stride) is DMA'd by the
//   Tensor Data Mover (tensor_load_to_lds, TENSORcnt), double-buffered.
// ---------------------------------------------------------------------------

typedef __attribute__((ext_vector_type(2))) float v2f;
typedef __attribute__((ext_vector_type(8))) float v8f;
typedef __attribute__((ext_vector_type(4))) unsigned u32x4;
typedef __attribute__((ext_vector_type(8))) unsigned u32x8;

#define LATENT 128
#define BATCH  256
#define TSTEPS 2000
#define LEAK   0.2f
#define RATE   0.001f
#define INVB   (1.0f / 256.0f)

// LDS partition (in floats)
#define SH_OFF   0                          // h        : 256*128
#define SW_OFF   (SH_OFF  + BATCH * LATENT) // W        : 128*128
#define SHN_OFF  (SW_OFF  + LATENT * LATENT)// h' stage : 32*128 (one macro-tile)
#define SX_OFF   (SHN_OFF + 32 * LATENT)    // x stage  : 2 * 256*2 (double buf)
#define SDEC_OFF (SX_OFF  + 2 * BATCH * 2)  // decay    : 2 * 128  (double buf)
#define SMEM_FLOATS (SDEC_OFF + 2 * LATENT) // = 54528 floats = 218112 bytes

__device__ __forceinline__ v8f wmma_f32x4(v2f a, v2f b, v8f c) {
  // D = A(16x4,f32) * B(4x16,f32) + C(16x16,f32)
  return __builtin_amdgcn_wmma_f32_16x16x4_f32(
      /*neg_a=*/false, a, /*neg_b=*/false, b,
      /*c_mod=*/(short)0, c, /*reuse_a=*/false, /*reuse_b=*/false);
}

// TDM: DMA x[:, t, :] (256 rows of 8 B, row stride 4000 floats) into LDS.
// D# group0/group1 packed per CDNA5 ISA ch.8 (CDNA5 ISA).
__device__ __forceinline__ void tdm_stage_x(const float* __restrict__ xg,
                                            int t, const float* ldsDst) {
  unsigned long long ga =
      (unsigned long long)(const void*)xg + (unsigned long long)t * 8ull;
  unsigned ldsOff = (unsigned)(unsigned long long)(const void*)ldsDst;

  u32x4 g0;
  g0[0] = 0x1u;                                   // count=1, no gather/restore
  g0[1] = ldsOff;                                 // lds_addr [63:32]
  g0[2] = (unsigned)(ga & 0xFFFFFFFFu);           // global_addr [95:64]
  g0[3] = (unsigned)((ga >> 32) & 0x01FFFFFFu)    // global_addr [120:96]
          | 0x80000000u;                          // type=2 ("image") [127:126]

  u32x8 g1;
  g1[0] = 0x00020000u;            // wg_mask=0, data_size=2 (4 B)
  g1[1] = (4000u & 0xFFFFu) << 16;// tensor_dim0[15:0] = T*2 = 4000
  g1[2] = (4000u >> 16) | (256u << 16); // tensor_dim0 hi | tensor_dim1=256 lo
  g1[3] = (2u << 16);             // tensor_dim1 hi=0 | tile_dim0=2
  g1[4] = 256u;                   // tile_dim1=256, tile_dim2=0 (unused)
  g1[5] = 4000u;                  // tensor_dim0_stride[31:0] = 4000 elements
  g1[6] = 0u;                     // stride hi=0, tensor_dim1_stride lo=0
  g1[7] = 0u;                     // tensor_dim1_stride hi=0

  asm volatile("tensor_load_to_lds %0, %1" :: "s"(g0), "s"(g1) : "memory");
}

extern "C" __global__ __launch_bounds__(512, 1)
void hebbian_chimera_kernel(const float* __restrict__ xg,     // [256,2000,2]
                            const float* __restrict__ Wres,   // [128,128]
                            const float* __restrict__ Win,    // [2,128]
                            const float* __restrict__ biasg,  // [128]
                            const int*   __restrict__ plast_p,
                            float* __restrict__ outp)         // [256,2000,128]
{
  extern __shared__ float smem[];
  float* sH   = smem + SH_OFF;
  float* sW   = smem + SW_OFF;
  float* sHN  = smem + SHN_OFF;
  float* sX   = smem + SX_OFF;
  float* sDec = smem + SDEC_OFF;

  const int tid  = threadIdx.x;
  const int w    = tid >> 5;        // wave 0..15
  const int l    = tid & 31;        // lane 0..31
  const int m16  = l & 15;          // M/N index within a 16x16 tile
  const int hi   = l >> 4;          // lane half (0/1)
  const int koff = hi << 1;         // K offsets 0,1 | 2,3 (ISA A/B layout)
  const int rt   = w >> 3;          // drive row-tile within a 32-row macro
  const int ct   = w & 7;           // drive col-tile
  const int col  = (ct << 4) + m16; // fixed output column per lane

  const int   plast = plast_p[0];
  const float biasv = biasg[col];
  const float wi0   = Win[col];
  const float wi1   = Win[LATENT + col];

  // per-lane column base: keeps DS offsets <64KB so they fold into ds_load imm
  const float* wCol = sW + col;

  // ---- init: h=0, W<-W_res, decay=0, DMA x[:,0,:] ----
  for (int i = tid; i < BATCH * LATENT; i += 512)  sH[i] = 0.0f;
  for (int i = tid; i < LATENT * LATENT; i += 512) sW[i] = Wres[i];
  if (tid < 2 * LATENT) sDec[tid] = 0.0f;
  if (w == 0) {
    tdm_stage_x(xg, 0, sX);               // buffer 0
    __builtin_amdgcn_s_wait_tensorcnt(0);
  }

  // per-wave Hebbian f32 accumulators: 4 fixed 16x16 tiles, resident in VGPRs
  v8f hAcc[4];
#pragma unroll
  for (int q = 0; q < 4; ++q) hAcc[q] = (v8f){0, 0, 0, 0, 0, 0, 0, 0};

  __syncthreads();

  for (int t = 0; t < TSTEPS; ++t) {
    const int cur = t & 1, nxt = cur ^ 1;

    for (int mac = 0; mac < 8; ++mac) {
      const int mB = mac << 5;  // first batch row of this 32-row macro-tile

      // ---- drive GEMM: C[16x16] += h[rows, k] * W[k, cols], K=128 ----
      v8f acc = (v8f){0, 0, 0, 0, 0, 0, 0, 0};
      const float* hRow = sH + (size_t)(mB + rt * 16 + m16) * LATENT;
#pragma unroll 8
      for (int k0 = 0; k0 < LATENT; k0 += 4) {
        v2f a, b;
        a.x = hRow[k0 + koff];                 // A: M=lane%16, K=koff..
        a.y = hRow[k0 + koff + 1];
        b.x = wCol[(k0 + koff) * LATENT];      // B: N=lane%16, K=koff..
        b.y = wCol[(k0 + koff + 1) * LATENT];
        acc = wmma_f32x4(a, b, acc);
      }

      // ---- epilogue: +u@W_in+bias, tanh, leak; stage h', emit out, decay ----
      float dsq = 0.0f;
#pragma unroll
      for (int r = 0; r < 8; ++r) {
        const int rowL = rt * 16 + (hi << 3) + r;   // C/D: M = r + 8*hi
        const int gb   = mB + rowL;
        const float u0 = sX[cur * 512 + gb * 2 + 0];
        const float u1 = sX[cur * 512 + gb * 2 + 1];
        const float d  = acc[r] + u0 * wi0 + u1 * wi1 + biasv;
        const float ho = sH[(size_t)gb * LATENT + col];
        const float hn = (1.0f - LEAK) * ho + LEAK * tanhf(d);
        sHN[rowL * LATENT + col] = hn;
        outp[((size_t)gb * TSTEPS + t) * LATENT + col] = hn;
        dsq += hn * hn;
      }
      atomicAdd(&sDec[cur * LATENT + col], dsq);    // ds_add_f32 (no-return)
      __syncthreads();  // B1: h' staged, drive reads of sW/sH retired

      // ---- hebb accumulation: hebb[i,j] += h'[b,i]*h[b,j], K=32 batch ----
#pragma unroll
      for (int q = 0; q < 4; ++q) {
        const int idx = (w << 2) + q;
        const int ti = idx >> 3, tj = idx & 7;
        const float* aCol = sHN + (ti << 4) + m16;                    // h'^T
        const float* bCol = sH + (size_t)mB * LATENT + (tj << 4) + m16; // h_old
        v8f hc = hAcc[q];
#pragma unroll
        for (int kb = 0; kb < 32; kb += 4) {
          v2f a, b;
          a.x = aCol[(kb + koff) * LATENT];
          a.y = aCol[(kb + koff + 1) * LATENT];
          b.x = bCol[(kb + koff) * LATENT];
          b.y = bCol[(kb + koff + 1) * LATENT];
          hc = wmma_f32x4(a, b, hc);
        }
        hAcc[q] = hc;
      }

      if (mac == 7) {
        // ---- W update (no wave reads sW between macro-7 B1 and step end) ----
        if (plast) {
#pragma unroll
          for (int q = 0; q < 4; ++q) {
            const int idx = (w << 2) + q;
            const int ti = idx >> 3, tj = idx & 7;
#pragma unroll
            for (int r = 0; r < 8; ++r) {
              const int i = ti * 16 + (hi << 3) + r;
              const int j = tj * 16 + m16;
              float wv = sW[i * LATENT + j];
              wv += RATE * (hAcc[q][r] * INVB -
                            (sDec[cur * LATENT + i] * INVB) * wv);
              sW[i * LATENT + j] = wv;
            }
          }
        }
#pragma unroll
        for (int q = 0; q < 4; ++q) hAcc[q] = (v8f){0, 0, 0, 0, 0, 0, 0, 0};
        if (tid < LATENT) sDec[nxt * LATENT + tid] = 0.0f;
        if (t + 1 < TSTEPS && w == 0)               // TDM: stage x[:,t+1,:]
          tdm_stage_x(xg, t + 1, sX + nxt * 512);
        if (t + 2 < TSTEPS && tid < BATCH)          // global_prefetch_b8
          __builtin_prefetch(&xg[(size_t)tid * (TSTEPS * 2) +
                                 (size_t)(t + 2) * 2], 0, 0);
      }
      __syncthreads();  // B2: hebb/update reads of sH/sHN/sDec retired

      // ---- commit h <- h' for this macro (4096 floats, 8 per thread) ----
      {
        const int off = tid * 8;
        float4* dst = (float4*)(sH + (size_t)mB * LATENT + off);
        const float4* src = (const float4*)(sHN + off);
        dst[0] = src[0];
        dst[1] = src[1];
      }
      if (mac == 7 && w == 0)
        __builtin_amdgcn_s_wait_tensorcnt(0);  // retire x[:,t+1,:] DMA
      __syncthreads();  // B3: h committed; sHN (and staged x) published
    }
  }
}

extern "C" void kernel_launch(void* const* d_in, const int* in_sizes, int n_in,
                              void* d_out, int out_size, void* d_ws, size_t ws_size,
                              hipStream_t stream) {
  const float* x     = (const float*)d_in[0];  // [256,2000,2]
  const float* Wres  = (const float*)d_in[1];  // [128,128]
  const float* Win   = (const float*)d_in[2];  // [2,128]
  const float* bias  = (const float*)d_in[3];  // [128]
  const int*   plast = (const int*)d_in[4];    // scalar
  float* out = (float*)d_out;                  // [256,2000,128]

  const size_t smem_bytes = (size_t)SMEM_FLOATS * sizeof(float); // 218112 B
  (void)hipFuncSetAttribute((const void*)hebbian_chimera_kernel,
                            hipFuncAttributeMaxDynamicSharedMemorySize,
                            (int)smem_bytes);
  hebbian_chimera_kernel<<<dim3(1), dim3(512), smem_bytes, stream>>>(
      x, Wres, Win, bias, plast, out);
}